// Decoder_63591285785155
// MI455X (gfx1250) — compile-verified
//
#include <hip/hip_runtime.h>
#include <hip/hip_bf16.h>
#include <math.h>

// ---------------------------------------------------------------------------
// Problem constants (from reference): B=64, S=2048, E=H=512, V=32000
// ---------------------------------------------------------------------------
#define Bn   64
#define Sn   2048
#define Hn   512
#define En   512
#define Vn   32000

typedef float v2f __attribute__((ext_vector_type(2)));
typedef float v8f __attribute__((ext_vector_type(8)));

// workspace layout (float offsets)
#define WS_X       0                       // x = relu(emb[idx])      [64][512]
#define WS_GI      32768                   // gi                      [64][1536]
#define WS_GH      131072                  // gh                      [64][1536]
#define WS_ACT     229376                  // [h | context]           [64][1024]
#define WS_U       294912                  // u = h @ W_attn          [64][512]
#define WS_C       327680                  // c[b] = b_attn . h[b]    [64]
#define WS_ALPHA   327744                  // alpha/scores            [64][2048]
#define WS_PART    458816                  // context partials        [64][16][512]
#define WS_END     983104                  // ~3.93 MB of floats

static __device__ __forceinline__ float waveReduceSum(float v) {
#pragma unroll
  for (int off = 16; off > 0; off >>= 1)
    v += __shfl_xor(v, off, 32);
  return v;
}

// ---------------------------------------------------------------------------
// 1) x[b,e] = relu(emb[idx[b], e])
// ---------------------------------------------------------------------------
__global__ void embed_relu_kernel(const int* __restrict__ idx,
                                  const float* __restrict__ emb,
                                  float* __restrict__ x) {
  int tid = blockIdx.x * blockDim.x + threadIdx.x;   // 64*512 threads
  int b = tid >> 9, e = tid & 511;
  float v = emb[(size_t)idx[b] * En + e];
  x[tid] = v > 0.0f ? v : 0.0f;
}

// ---------------------------------------------------------------------------
// Generic fp32 WMMA GEMM:  out[m,n] = sum_k A[m, k] * W(n, k) + bias[n]
//   A: row-major, leading dim lda, M fixed = 64 (4 waves x 16 rows)
//   W element (n,k) at W[n*sN + k*sK]  (sK==1 -> A@W^T; sN==1 -> A@W)
//   One 16x16 D-tile per wave via V_WMMA_F32_16X16X4_F32, K-step of 4.
// ISA layouts (CDNA5 ISA §7.12.2):
//   A 16x4 : lane&15 = M row; lanes>=16 hold K=2,3 (v0:K0/K2, v1:K1/K3)
//   B 4x16 : lane&15 = N col; lanes>=16 hold K=2,3 rows
//   D      : v[j] = (M = j + 8*(lane>=16), N = lane&15)
// ---------------------------------------------------------------------------
__global__ void __launch_bounds__(128)
wmma_gemm_f32(const float* __restrict__ A, int lda,
              const float* __restrict__ W, int sN, int sK,
              const float* __restrict__ bias,
              float* __restrict__ out, int ldo, int K) {
  const int lane  = threadIdx.x & 31;
  const int wave  = threadIdx.x >> 5;            // m-tile 0..3
  const int mrow  = wave * 16 + (lane & 15);
  const int ncol  = blockIdx.x * 16 + (lane & 15);
  const int khalf = (lane >> 4) << 1;            // 0 or 2
  const float* ap = A + (size_t)mrow * lda + khalf;

  v8f acc = {};
  if (sK == 1) {
    const float* wp = W + (size_t)ncol * sN + khalf;
    for (int k = 0; k < K; k += 4) {
      v2f a = *(const v2f*)(ap + k);
      v2f b = *(const v2f*)(wp + k);
      acc = __builtin_amdgcn_wmma_f32_16x16x4_f32(
          /*neg_a=*/false, a, /*neg_b=*/false, b,
          /*c_mod=*/(short)0, acc, /*reuse_a=*/false, /*reuse_b=*/false);
    }
  } else {
    for (int k = 0; k < K; k += 4) {
      v2f a = *(const v2f*)(ap + k);
      v2f b;
      b.x = W[(size_t)ncol * sN + (size_t)(k + khalf)     * sK];
      b.y = W[(size_t)ncol * sN + (size_t)(k + khalf + 1) * sK];
      acc = __builtin_amdgcn_wmma_f32_16x16x4_f32(
          false, a, false, b, (short)0, acc, false, false);
    }
  }
  const float bv = bias ? bias[ncol] : 0.0f;
  const int hi = lane >> 4;
#pragma unroll
  for (int j = 0; j < 8; ++j) {
    int mr = wave * 16 + j + 8 * hi;
    out[(size_t)mr * ldo + ncol] = acc[j] + bv;
  }
}

// ---------------------------------------------------------------------------
// 4) GRU gate combine (PyTorch layout r,z,n); writes h into act[:, :512]
//    and into d_out tail (dec_hidden).
// ---------------------------------------------------------------------------
__global__ void gru_combine_kernel(const float* __restrict__ gi,
                                   const float* __restrict__ gh,
                                   const float* __restrict__ h0,
                                   float* __restrict__ act,
                                   float* __restrict__ dec_hidden_out) {
  int b = blockIdx.x, j = threadIdx.x;             // 64 blocks x 512
  const float* gib = gi + (size_t)b * 3 * Hn;
  const float* ghb = gh + (size_t)b * 3 * Hn;
  float r = 1.0f / (1.0f + expf(-(gib[j]          + ghb[j])));
  float z = 1.0f / (1.0f + expf(-(gib[Hn + j]     + ghb[Hn + j])));
  float n = tanhf(gib[2 * Hn + j] + r * ghb[2 * Hn + j]);
  float hv = (1.0f - z) * n + z * h0[(size_t)b * Hn + j];
  act[(size_t)b * 1024 + j] = hv;
  dec_hidden_out[(size_t)b * Hn + j] = hv;
}

// ---------------------------------------------------------------------------
// 6) c[b] = b_attn . h[b]   (one wave per b)
// ---------------------------------------------------------------------------
__global__ void cvec_kernel(const float* __restrict__ b_attn,
                            const float* __restrict__ act,
                            float* __restrict__ c) {
  int b = blockIdx.x, lane = threadIdx.x;          // 64 blocks x 32
  float acc = 0.0f;
  for (int k = lane; k < Hn; k += 32)
    acc += b_attn[k] * act[(size_t)b * 1024 + k];
  acc = waveReduceSum(acc);
  if (lane == 0) c[b] = acc;
}

// ---------------------------------------------------------------------------
// 7) alpha[b,s] = enc[s,b,:] . u[b,:] + c[b]   (one wave per (s,b) pair)
//    Pure-bandwidth pass #1 over enc (268 MB).
// ---------------------------------------------------------------------------
__global__ void attn_alpha_kernel(const float* __restrict__ enc,
                                  const float* __restrict__ u,
                                  const float* __restrict__ c,
                                  float* __restrict__ alpha) {
  int lane = threadIdx.x & 31;
  int w = blockIdx.x * (blockDim.x >> 5) + (threadIdx.x >> 5);
  int s = w >> 6, b = w & 63;
  const float4* ep = (const float4*)(enc + ((size_t)s * Bn + b) * Hn);
  const float4* up = (const float4*)(u + (size_t)b * Hn);
  float acc = 0.0f;
#pragma unroll
  for (int i = 0; i < 4; ++i) {
    float4 e4 = ep[lane + i * 32];
    float4 u4 = up[lane + i * 32];
    acc += e4.x * u4.x + e4.y * u4.y + e4.z * u4.z + e4.w * u4.w;
  }
  acc = waveReduceSum(acc);
  if (lane == 0) alpha[(size_t)b * Sn + s] = acc + c[b];
}

// ---------------------------------------------------------------------------
// 8) softmax over S per row, in place (64 rows x 2048)
// ---------------------------------------------------------------------------
__global__ void softmax_rows_kernel(float* __restrict__ a) {
  __shared__ float red[256];
  int b = blockIdx.x, t = threadIdx.x;
  float* row = a + (size_t)b * Sn;
  float m = -3.0e38f;
  for (int i = t; i < Sn; i += 256) m = fmaxf(m, row[i]);
  red[t] = m; __syncthreads();
  for (int o = 128; o > 0; o >>= 1) {
    if (t < o) red[t] = fmaxf(red[t], red[t + o]);
    __syncthreads();
  }
  m = red[0]; __syncthreads();
  float s = 0.0f;
  for (int i = t; i < Sn; i += 256) s += expf(row[i] - m);
  red[t] = s; __syncthreads();
  for (int o = 128; o > 0; o >>= 1) {
    if (t < o) red[t] += red[t + o];
    __syncthreads();
  }
  float inv = 1.0f / red[0];
  for (int i = t; i < Sn; i += 256) row[i] = expf(row[i] - m) * inv;
}

// ---------------------------------------------------------------------------
// 9) context partials over 16 s-partitions. Bandwidth pass #2 over enc.
// ---------------------------------------------------------------------------
__global__ void ctx_partial_kernel(const float* __restrict__ enc,
                                   const float* __restrict__ scores,
                                   float* __restrict__ part) {
  int sp = blockIdx.x, b = blockIdx.y, t = threadIdx.x;  // (16,64) x 512
  float acc = 0.0f;
  for (int s = sp * 128; s < sp * 128 + 128; ++s)
    acc += scores[(size_t)b * Sn + s] * enc[((size_t)s * Bn + b) * Hn + t];
  part[((size_t)b * 16 + sp) * Hn + t] = acc;
}

// 10) reduce partials into act[:, 512:1024]
__global__ void ctx_reduce_kernel(const float* __restrict__ part,
                                  float* __restrict__ act) {
  int b = blockIdx.x, t = threadIdx.x;                   // 64 x 512
  float acc = 0.0f;
#pragma unroll
  for (int sp = 0; sp < 16; ++sp)
    acc += part[((size_t)b * 16 + sp) * Hn + t];
  act[(size_t)b * 1024 + 512 + t] = acc;
}

// ---------------------------------------------------------------------------
// 12) in-place log-softmax over 32000 per row of d_out
// ---------------------------------------------------------------------------
__global__ void log_softmax_kernel(float* __restrict__ out) {
  __shared__ float red[256];
  int b = blockIdx.x, t = threadIdx.x;
  float* row = out + (size_t)b * Vn;
  float m = -3.0e38f;
  for (int i = t; i < Vn; i += 256) m = fmaxf(m, row[i]);
  red[t] = m; __syncthreads();
  for (int o = 128; o > 0; o >>= 1) {
    if (t < o) red[t] = fmaxf(red[t], red[t + o]);
    __syncthreads();
  }
  m = red[0]; __syncthreads();
  float s = 0.0f;
  for (int i = t; i < Vn; i += 256) s += expf(row[i] - m);
  red[t] = s; __syncthreads();
  for (int o = 128; o > 0; o >>= 1) {
    if (t < o) red[t] += red[t + o];
    __syncthreads();
  }
  float lse = logf(red[0]) + m;
  for (int i = t; i < Vn; i += 256) row[i] = row[i] - lse;
}

// ---------------------------------------------------------------------------
extern "C" void kernel_launch(void* const* d_in, const int* in_sizes, int n_in,
                              void* d_out, int out_size, void* d_ws, size_t ws_size,
                              hipStream_t stream) {
  const int*   idx     = (const int*)  d_in[0];
  const float* enc     = (const float*)d_in[1];
  const float* h0      = (const float*)d_in[2];
  const float* emb     = (const float*)d_in[3];
  const float* W_ih    = (const float*)d_in[4];
  const float* W_hh    = (const float*)d_in[5];
  const float* b_ih    = (const float*)d_in[6];
  const float* b_hh    = (const float*)d_in[7];
  const float* W_attn  = (const float*)d_in[8];
  const float* b_attn  = (const float*)d_in[9];
  const float* W_cls   = (const float*)d_in[10];
  const float* b_cls   = (const float*)d_in[11];
  float* out = (float*)d_out;                       // [64][32000] then [64][512]
  float* ws  = (float*)d_ws;

  float* x     = ws + WS_X;
  float* gi    = ws + WS_GI;
  float* gh    = ws + WS_GH;
  float* act   = ws + WS_ACT;
  float* u     = ws + WS_U;
  float* c     = ws + WS_C;
  float* alpha = ws + WS_ALPHA;
  float* part  = ws + WS_PART;
  float* dec_hidden = out + (size_t)Bn * Vn;

  // 1) embedding + ReLU
  embed_relu_kernel<<<(Bn * En) / 256, 256, 0, stream>>>(idx, emb, x);

  // 2,3) GRU input/hidden GEMMs (fp32 WMMA):  [64,512] @ [1536,512]^T
  wmma_gemm_f32<<<(3 * Hn) / 16, 128, 0, stream>>>(x, En, W_ih, En, 1, b_ih, gi, 3 * Hn, En);
  wmma_gemm_f32<<<(3 * Hn) / 16, 128, 0, stream>>>(h0, Hn, W_hh, Hn, 1, b_hh, gh, 3 * Hn, Hn);

  // 4) gates -> h (into act[:, :512] and dec_hidden output)
  gru_combine_kernel<<<Bn, Hn, 0, stream>>>(gi, gh, h0, act, dec_hidden);

  // 5) u = h @ W_attn  (W non-transposed: element (n,k) at W_attn[k*H + n])
  wmma_gemm_f32<<<Hn / 16, 128, 0, stream>>>(act, 1024, W_attn, 1, Hn, nullptr, u, Hn, Hn);

  // 6) c[b] = b_attn . h[b]
  cvec_kernel<<<Bn, 32, 0, stream>>>(b_attn, act, c);

  // 7) alpha[b,s] = enc[s,b].u[b] + c[b]   (bandwidth pass #1)
  attn_alpha_kernel<<<(Sn * Bn) / 8, 256, 0, stream>>>(enc, u, c, alpha);

  // 8) softmax rows
  softmax_rows_kernel<<<Bn, 256, 0, stream>>>(alpha);

  // 9,10) context (bandwidth pass #2) -> act[:, 512:]
  ctx_partial_kernel<<<dim3(16, Bn), Hn, 0, stream>>>(enc, alpha, part);
  ctx_reduce_kernel<<<Bn, Hn, 0, stream>>>(part, act);

  // 11) logits = [h|context] @ W_cls^T + b_cls  -> directly into d_out
  wmma_gemm_f32<<<Vn / 16, 128, 0, stream>>>(act, 1024, W_cls, 2 * Hn, 1, b_cls, out, Vn, 2 * Hn);

  // 12) in-place log-softmax
  log_softmax_kernel<<<Bn, 256, 0, stream>>>(out);
}